// DepBinaryClassifier_2010044694684
// MI455X (gfx1250) — compile-verified
//
#include <hip/hip_runtime.h>
#include <hip/hip_bf16.h>
#include <math.h>

// ---------------- model constants ----------------
#define BB   512
#define SS   150
#define FIN  75
#define EE   128
#define DD   256
#define NHM  4
#define DHM  64
#define FFN_ 192
#define MM   (BB*SS)          // 76800 rows, multiple of 16
#define BH   (BB*NHM)         // 2048

typedef float v2f __attribute__((ext_vector_type(2)));
typedef float v8f __attribute__((ext_vector_type(8)));

// fp32 WMMA: D(16x16 f32) = A(16x4 f32) * B(4x16 f32) + C
// A layout: lane<16 -> m=lane, elems {K=0,K=1}; lane>=16 -> m=lane-16, elems {K=2,K=3}
// B layout: lane<16 -> n=lane, elems {K=0,K=1}; lane>=16 -> n=lane-16, elems {K=2,K=3}
// C/D layout: elem j -> row j + 8*(lane>>4), col lane&15
static __device__ inline v8f wmma4(v2f a, v2f b, v8f c) {
  return __builtin_amdgcn_wmma_f32_16x16x4_f32(false, a, false, b, (short)0, c,
                                               false, false);
}

static __device__ inline float sigmoidf_(float x) { return 1.f / (1.f + __expf(-x)); }
static __device__ inline float siluf_(float x) { return x * sigmoidf_(x); }
static __device__ inline float logsigf_(float x) {
  return fminf(x, 0.f) - log1pf(__expf(-fabsf(x)));
}

// ------- generic fp32 WMMA GEMM: C = (accum?C:0) + A@W + bias ------------------
// Each wave computes a 16(M) x 32(N) tile (two 16x16 WMMA tiles sharing A).
// Main K-loop is branchless/unguarded; single branchless tail handles K%4.
__global__ void __launch_bounds__(128)
gemm_wmma_f32(const float* __restrict__ A, int lda,
              const float* __restrict__ W, int ldb,
              const float* __restrict__ bias,
              float* __restrict__ C, int ldc,
              int M, int N, int K, int accum) {
  int w = blockIdx.x * (blockDim.x >> 5) + (threadIdx.x >> 5);
  int lane = threadIdx.x & 31;
  int nt2 = N >> 5;  // number of 32-wide column tiles
  int mt = w / nt2, nt = w % nt2;
  if (mt >= (M >> 4)) return;
  int m0 = mt << 4, n0 = nt << 5;
  int mrow = lane & 15;
  int kOff = (lane >> 4) << 1;

  v8f acc0 = {}, acc1 = {};
  const float* Arow = A + (size_t)(m0 + mrow) * lda + kOff;
  const float* W0 = W + (size_t)kOff * ldb + n0 + mrow;
  const float* W1 = W0 + 16;
  int K4 = K & ~3;

  const float* ap = Arow;
  const float* wp = W0;
#pragma unroll 2
  for (int kk = 0; kk < K4; kk += 4) {
    v2f a, b0, b1;
    a.x = ap[0];
    a.y = ap[1];
    b0.x = wp[0];
    b0.y = wp[ldb];
    b1.x = wp[16];
    b1.y = wp[ldb + 16];
    acc0 = wmma4(a, b0, acc0);
    acc1 = wmma4(a, b1, acc1);
    ap += 4;
    wp += (size_t)4 * ldb;
  }
  if (K4 < K) {  // branchless tail: select address (always valid) + zero via select
    int k0 = K4 + kOff, k1 = k0 + 1;
    bool i0 = k0 < K, i1 = k1 < K;
    size_t a0i = i0 ? (size_t)K4 : 0, a1i = i1 ? (size_t)K4 + 1 : 0;
    size_t w0i = i0 ? (size_t)K4 * ldb : 0, w1i = i1 ? ((size_t)K4 + 1) * ldb : 0;
    v2f a, b0, b1;
    float t;
    t = Arow[a0i]; a.x = i0 ? t : 0.f;
    t = Arow[a1i]; a.y = i1 ? t : 0.f;
    t = W0[w0i];   b0.x = i0 ? t : 0.f;
    t = W0[w1i];   b0.y = i1 ? t : 0.f;
    t = W1[w0i];   b1.x = i0 ? t : 0.f;
    t = W1[w1i];   b1.y = i1 ? t : 0.f;
    acc0 = wmma4(a, b0, acc0);
    acc1 = wmma4(a, b1, acc1);
  }

  int col = n0 + mrow;
  float bv0 = bias ? bias[col] : 0.f;
  float bv1 = bias ? bias[col + 16] : 0.f;
  int rbase = m0 + ((lane >> 4) << 3);
  if (accum) {
#pragma unroll
    for (int j = 0; j < 8; ++j) {
      size_t idx = (size_t)(rbase + j) * ldc + col;
      C[idx] += acc0[j] + bv0;
      C[idx + 16] += acc1[j] + bv1;
    }
  } else {
#pragma unroll
    for (int j = 0; j < 8; ++j) {
      size_t idx = (size_t)(rbase + j) * ldc + col;
      C[idx] = acc0[j] + bv0;
      C[idx + 16] = acc1[j] + bv1;
    }
  }
}

// ---------------- layernorm over last dim = 128 (no bias) ----------------------
__global__ void __launch_bounds__(256)
layernorm128(const float* __restrict__ x, const float* __restrict__ w,
             float* __restrict__ y, int M) {
  int wid = blockIdx.x * (blockDim.x >> 5) + (threadIdx.x >> 5);
  if (wid >= M) return;
  int lane = threadIdx.x & 31;
  const float* xr = x + (size_t)wid * EE;
  float v[4], s = 0.f;
#pragma unroll
  for (int i = 0; i < 4; ++i) { v[i] = xr[lane + 32 * i]; s += v[i]; }
#pragma unroll
  for (int o = 16; o >= 1; o >>= 1) s += __shfl_xor(s, o);
  float mu = s * (1.f / 128.f);
  float vs = 0.f;
#pragma unroll
  for (int i = 0; i < 4; ++i) { v[i] -= mu; vs += v[i] * v[i]; }
#pragma unroll
  for (int o = 16; o >= 1; o >>= 1) vs += __shfl_xor(vs, o);
  float rstd = rsqrtf(vs * (1.f / 128.f) + 1e-5f);
  float* yr = y + (size_t)wid * EE;
#pragma unroll
  for (int i = 0; i < 4; ++i) yr[lane + 32 * i] = v[i] * rstd * w[lane + 32 * i];
}

// ---------------- depthwise causal conv (K=4) + SiLU (branchless taps) ---------
__global__ void convsilu_k(const float* __restrict__ x, int xstride,
                           const float* __restrict__ w, const float* __restrict__ bias,
                           float* __restrict__ y, int C, int total) {
  int idx = blockIdx.x * blockDim.x + threadIdx.x;
  if (idx >= total) return;
  int c = idx % C;
  int row = idx / C;
  int b = row / SS, s = row % SS;
  float acc = bias[c];
#pragma unroll
  for (int j = 0; j < 4; ++j) {
    int t = s - 3 + j;
    int tc = t < 0 ? 0 : t;
    float xv = x[((size_t)(b * SS + tc)) * xstride + c];
    float wv = (t >= 0) ? w[c * 4 + j] : 0.f;
    acc += wv * xv;
  }
  y[(size_t)row * C + c] = siluf_(acc);
}

// ---------------- block-diagonal 4x4 headwise (mLSTM q/k/v) --------------------
__global__ void headwise4_k(const float* __restrict__ x, int xstride,
                            const float* __restrict__ w, const float* __restrict__ bias,
                            float* __restrict__ y, int total) {
  int idx = blockIdx.x * blockDim.x + threadIdx.x;
  if (idx >= total) return;
  int c = idx & 255;
  int row = idx >> 8;
  int nb = c >> 2, o = c & 3;
  const float* wr = w + ((size_t)nb * 4 + o) * 4;
  const float* xr = x + (size_t)row * xstride + nb * 4;
  y[(size_t)row * DD + c] =
      bias[c] + wr[0] * xr[0] + wr[1] * xr[1] + wr[2] * xr[2] + wr[3] * xr[3];
}

// ---------------- block-diagonal 32x32 headwise (sLSTM gates, no bias) ---------
__global__ void headwise32_k(const float* __restrict__ x,
                             const float* __restrict__ w,
                             float* __restrict__ y, int total) {
  int idx = blockIdx.x * blockDim.x + threadIdx.x;
  if (idx >= total) return;
  int c = idx & 127;
  int row = idx >> 7;
  int h = c >> 5, o = c & 31;
  const float* wr = w + ((size_t)h * 32 + o) * 32;
  const float* xr = x + (size_t)row * EE + h * 32;
  float s = 0.f;
#pragma unroll
  for (int d = 0; d < 32; ++d) s += wr[d] * xr[d];
  y[(size_t)row * EE + c] = s;
}

// ------- ig/fg gate projections (K=768 -> 4+4), wave per row, 3 flat loops -----
__global__ void __launch_bounds__(256)
igfg_k(const float* __restrict__ q, const float* __restrict__ k,
       const float* __restrict__ v, const float* __restrict__ wi,
       const float* __restrict__ bi, const float* __restrict__ wf,
       const float* __restrict__ bf, float* __restrict__ ig, float* __restrict__ fg) {
  int wid = blockIdx.x * (blockDim.x >> 5) + (threadIdx.x >> 5);
  if (wid >= MM) return;
  int lane = threadIdx.x & 31;
  int b = wid / SS, s = wid % SS;
  size_t r = (size_t)wid * DD;
  float aI[4] = {0, 0, 0, 0}, aF[4] = {0, 0, 0, 0};
#pragma unroll 2
  for (int kk = lane; kk < DD; kk += 32) {
    float xv = q[r + kk];
    const float* wIr = wi + (size_t)kk * 4;
    const float* wFr = wf + (size_t)kk * 4;
#pragma unroll
    for (int o = 0; o < 4; ++o) { aI[o] += xv * wIr[o]; aF[o] += xv * wFr[o]; }
  }
#pragma unroll 2
  for (int kk = lane; kk < DD; kk += 32) {
    float xv = k[r + kk];
    const float* wIr = wi + (size_t)(kk + DD) * 4;
    const float* wFr = wf + (size_t)(kk + DD) * 4;
#pragma unroll
    for (int o = 0; o < 4; ++o) { aI[o] += xv * wIr[o]; aF[o] += xv * wFr[o]; }
  }
#pragma unroll 2
  for (int kk = lane; kk < DD; kk += 32) {
    float xv = v[r + kk];
    const float* wIr = wi + (size_t)(kk + 2 * DD) * 4;
    const float* wFr = wf + (size_t)(kk + 2 * DD) * 4;
#pragma unroll
    for (int o = 0; o < 4; ++o) { aI[o] += xv * wIr[o]; aF[o] += xv * wFr[o]; }
  }
#pragma unroll
  for (int o = 0; o < 4; ++o)
#pragma unroll
    for (int off = 16; off >= 1; off >>= 1) {
      aI[o] += __shfl_xor(aI[o], off);
      aF[o] += __shfl_xor(aF[o], off);
    }
  if (lane == 0) {
#pragma unroll
    for (int o = 0; o < 4; ++o) {
      ig[((size_t)(b * NHM + o)) * SS + s] = aI[o] + bi[o];
      fg[((size_t)(b * NHM + o)) * SS + s] = aF[o] + bf[o];
    }
  }
}

// ------ per-(b,h) scan: lfc[t+1], prefix-max(ig-lfc), exp(-maxD) ---------------
__global__ void mlstm_scan_k(const float* __restrict__ ig, const float* __restrict__ fg,
                             float* __restrict__ lf, float* __restrict__ pm,
                             float* __restrict__ em) {
  int bh = blockIdx.x * blockDim.x + threadIdx.x;
  if (bh >= BH) return;
  size_t base = (size_t)bh * SS;
  float lfc = 0.f, p = -1e30f;
  for (int t = 0; t < SS; ++t) {
    lfc += logsigf_(fg[base + t]);
    float u = ig[base + t] - lfc;
    p = fmaxf(p, u);
    lf[base + t] = lfc;
    pm[base + t] = p;
    em[base + t] = __expf(-(lfc + p));  // exp(-maxD[s])
  }
}

// ---------------- mLSTM parallel attention, one workgroup per (b,h) ------------
// S padded to 160 (10 tiles). K,V tiles in LDS (row stride 68, conflict-free).
__global__ void __launch_bounds__(128)
mlstm_attn_k(const float* __restrict__ Q, const float* __restrict__ Kg,
             const float* __restrict__ V, const float* __restrict__ lf,
             const float* __restrict__ igv, const float* __restrict__ pm,
             const float* __restrict__ em, float* __restrict__ hc) {
  const int ST = 160, RK = 68;
  extern __shared__ float sm[];
  float* sK = sm;                 // 160*68
  float* sV = sK + ST * RK;       // 160*68
  float* sLf = sV + ST * RK;      // 160
  float* sIg = sLf + ST;          // 160
  float* sPm = sIg + ST;          // 160
  float* sEm = sPm + ST;          // 160
  float* sC = sEm + ST;           // 4 waves * 16*17

  int bh = blockIdx.x;
  int b = bh >> 2, h = bh & 3;
  int tid = threadIdx.x, lane = tid & 31, wv = tid >> 5;

  for (int i = tid; i < ST * 64; i += blockDim.x) {
    int t = i >> 6, d = i & 63;
    int tc = t < SS ? t : 0;
    float msk = t < SS ? 1.f : 0.f;
    size_t g = ((size_t)(b * SS + tc)) * DD + h * DHM + d;
    sK[t * RK + d] = Kg[g] * msk;
    sV[t * RK + d] = V[g] * msk;
  }
  for (int t = tid; t < ST; t += blockDim.x) {
    bool in = t < SS;
    size_t g = (size_t)bh * SS + (in ? t : 0);
    float lfv = lf[g], igvv = igv[g], pmv = pm[g], emv = em[g];
    sLf[t] = in ? lfv : 0.f;
    sIg[t] = in ? igvv : -1e30f;  // padded t -> weight 0
    sPm[t] = in ? pmv : 0.f;
    sEm[t] = in ? emv : 0.f;
  }
  __syncthreads();

  int mrow = lane & 15;
  int grp = lane >> 4;
  int kOff = grp << 1;
  float* myC = sC + wv * 272;  // wave-private 16x17 scratch

  for (int qt = wv; qt < 10; qt += 4) {
    int s0 = qt << 4;
    // preload Q A-fragments, branchless: clamp row, fold validity into scale
    v2f qa[16];
    int qrow = s0 + mrow;
    float qscale = (qrow < SS) ? 0.125f : 0.f;  // 1/sqrt(DH)=1/8, or mask
    const float* Qr = Q + ((size_t)(b * SS + (qrow < SS ? qrow : 0))) * DD + h * DHM + kOff;
#pragma unroll
    for (int kk = 0; kk < 16; ++kk) {
      v2f a;
      a.x = Qr[kk * 4] * qscale;
      a.y = Qr[kk * 4 + 1] * qscale;
      qa[kk] = a;
    }
    float pmrow[8], nacc[8];
    int rbase = s0 + (grp << 3);
#pragma unroll
    for (int j = 0; j < 8; ++j) { pmrow[j] = sPm[rbase + j]; nacc[j] = 0.f; }
    v8f o0 = {}, o1 = {}, o2 = {}, o3 = {};

    for (int tt = 0; tt <= qt; ++tt) {
      int t0 = tt << 4;
      // S = Q @ K^T  (contraction over d=64, 16 chained f32 WMMAs)
      v8f sf = {};
#pragma unroll
      for (int kk = 0; kk < 16; ++kk) {
        const float* kp = sK + (t0 + mrow) * RK + kk * 4 + kOff;
        v2f bfr;
        bfr.x = kp[0];
        bfr.y = kp[1];
        sf = wmma4(qa[kk], bfr, sf);
      }
      // causal mask + stabilized decay weight; rowsum for norm; stash to LDS
      int t = t0 + mrow;
      float dlt = sIg[t] - sLf[t];
#pragma unroll
      for (int j = 0; j < 8; ++j) {
        int r = rbase + j;
        float wgt = (t <= r) ? __expf(dlt - pmrow[j]) : 0.f;
        float cv = sf[j] * wgt;
        float rs = cv;
        rs += __shfl_xor(rs, 1);
        rs += __shfl_xor(rs, 2);
        rs += __shfl_xor(rs, 4);
        rs += __shfl_xor(rs, 8);
        nacc[j] += rs;
        myC[(j + (grp << 3)) * 17 + mrow] = cv;
      }
      // O += C @ V  (contraction over t=16, 16 f32 WMMAs across 4 d-tiles)
#pragma unroll
      for (int kk2 = 0; kk2 < 4; ++kk2) {
        v2f a;
        a.x = myC[mrow * 17 + kk2 * 4 + kOff];
        a.y = myC[mrow * 17 + kk2 * 4 + kOff + 1];
        int tr = t0 + kk2 * 4 + kOff;
        const float* v0 = sV + tr * RK;
        const float* v1 = sV + (tr + 1) * RK;
        v2f b0; b0.x = v0[mrow];      b0.y = v1[mrow];      o0 = wmma4(a, b0, o0);
        v2f b1; b1.x = v0[16 + mrow]; b1.y = v1[16 + mrow]; o1 = wmma4(a, b1, o1);
        v2f b2; b2.x = v0[32 + mrow]; b2.y = v1[32 + mrow]; o2 = wmma4(a, b2, o2);
        v2f b3; b3.x = v0[48 + mrow]; b3.y = v1[48 + mrow]; o3 = wmma4(a, b3, o3);
      }
    }
    // norm = max(|sum C|, exp(-maxD)); out = O / (norm + 1e-6)
#pragma unroll
    for (int j = 0; j < 8; ++j) {
      int r = rbase + j;
      if (r < SS) {
        float nm = fmaxf(fabsf(nacc[j]), sEm[r]);
        float inv = 1.f / (nm + 1e-6f);
        size_t base = ((size_t)(b * SS + r) * NHM + h) * DHM + mrow;
        hc[base + 0] = o0[j] * inv;
        hc[base + 16] = o1[j] * inv;
        hc[base + 32] = o2[j] * inv;
        hc[base + 48] = o3[j] * inv;
      }
    }
  }
}

// ------- fused per-head groupnorm(DH=64) + skip + output gate ------------------
__global__ void __launch_bounds__(256)
fuse_hs_k(const float* __restrict__ hc, const float* __restrict__ xc,
          const float* __restrict__ up, const float* __restrict__ normw,
          const float* __restrict__ skip, float* __restrict__ hs) {
  int wid = blockIdx.x * (blockDim.x >> 5) + (threadIdx.x >> 5);
  if (wid >= MM * NHM) return;
  int lane = threadIdx.x & 31;
  int r = wid >> 2, h = wid & 3;
  size_t base = ((size_t)r * NHM + h) * DHM;
  float v0 = hc[base + lane], v1 = hc[base + 32 + lane];
  float s = v0 + v1;
#pragma unroll
  for (int o = 16; o >= 1; o >>= 1) s += __shfl_xor(s, o);
  float mu = s * (1.f / 64.f);
  v0 -= mu; v1 -= mu;
  float vs = v0 * v0 + v1 * v1;
#pragma unroll
  for (int o = 16; o >= 1; o >>= 1) vs += __shfl_xor(vs, o);
  float rstd = rsqrtf(vs * (1.f / 64.f) + 1e-5f);
#pragma unroll
  for (int half = 0; half < 2; ++half) {
    int d = lane + 32 * half;
    int c = h * DHM + d;
    float hn = (half ? v1 : v0) * rstd * normw[c];
    float zv = up[(size_t)r * (2 * DD) + DD + c];
    hs[(size_t)r * DD + c] = (hn + skip[c] * xc[(size_t)r * DD + c]) * siluf_(zv);
  }
}

// ---------------- sLSTM sequential recurrence: one wave per (b,h) --------------
__global__ void __launch_bounds__(256)
slstm_scan_k(const float* __restrict__ g0, const float* __restrict__ g1,
             const float* __restrict__ g2, const float* __restrict__ g3,
             const float* __restrict__ sr, const float* __restrict__ sb,
             float* __restrict__ yout) {
  int wid = blockIdx.x * (blockDim.x >> 5) + (threadIdx.x >> 5);
  if (wid >= BB * 4) return;
  int lane = threadIdx.x & 31;
  int b = wid >> 2, h = wid & 3;
  // recurrent weights for this head, column = lane: 128 VGPRs
  float4 Rr[32];
#pragma unroll
  for (int d = 0; d < 32; ++d) {
    Rr[d].x = sr[(((size_t)0 * 4 + h) * 32 + d) * 32 + lane];
    Rr[d].y = sr[(((size_t)1 * 4 + h) * 32 + d) * 32 + lane];
    Rr[d].z = sr[(((size_t)2 * 4 + h) * 32 + d) * 32 + lane];
    Rr[d].w = sr[(((size_t)3 * 4 + h) * 32 + d) * 32 + lane];
  }
  float bi = sb[(0 * 4 + h) * 32 + lane];
  float bf = sb[(1 * 4 + h) * 32 + lane];
  float bz = sb[(2 * 4 + h) * 32 + lane];
  float bo = sb[(3 * 4 + h) * 32 + lane];
  float y = 0.f, c = 0.f, n = 0.f, m = -1e30f;
  for (int s = 0; s < SS; ++s) {
    size_t base = (((size_t)(b * SS + s)) * 4 + h) * 32 + lane;
    float r0 = 0.f, r1 = 0.f, r2 = 0.f, r3 = 0.f;
#pragma unroll
    for (int d = 0; d < 32; ++d) {
      float yd = __shfl(y, d);
      r0 += yd * Rr[d].x;
      r1 += yd * Rr[d].y;
      r2 += yd * Rr[d].z;
      r3 += yd * Rr[d].w;
    }
    float ir = g0[base] + r0 + bi;
    float fr = g1[base] + r1 + bf;
    float zr = g2[base] + r2 + bz;
    float orr = g3[base] + r3 + bo;
    float lfm = m + logsigf_(fr);
    float mn = fmaxf(ir, lfm);
    float igt = __expf(ir - mn);
    float fgt = __expf(lfm - mn);
    c = fgt * c + igt * tanhf(zr);
    n = fgt * n + igt;
    y = sigmoidf_(orr) * c / n;
    m = mn;
    yout[base] = y;
  }
}

// ------- per-head groupnorm(DH=32) of sLSTM output, added into residual --------
__global__ void __launch_bounds__(256)
fuse_ys_k(const float* __restrict__ yout, const float* __restrict__ normw,
          float* __restrict__ hbuf) {
  int wid = blockIdx.x * (blockDim.x >> 5) + (threadIdx.x >> 5);
  if (wid >= MM * 4) return;
  int lane = threadIdx.x & 31;
  int r = wid >> 2, h = wid & 3;
  size_t base = ((size_t)r * 4 + h) * 32 + lane;
  float v = yout[base];
  float s = v;
#pragma unroll
  for (int o = 16; o >= 1; o >>= 1) s += __shfl_xor(s, o);
  float mu = s * (1.f / 32.f);
  v -= mu;
  float vs = v * v;
#pragma unroll
  for (int o = 16; o >= 1; o >>= 1) vs += __shfl_xor(vs, o);
  float rstd = rsqrtf(vs * (1.f / 32.f) + 1e-5f);
  hbuf[(size_t)r * EE + h * 32 + lane] += v * rstd * normw[h * 32 + lane];
}

// ---------------- exact GELU(gate) * up ----------------------------------------
__global__ void gelumul_k(const float* __restrict__ up2, float* __restrict__ act,
                          int total) {
  int idx = blockIdx.x * blockDim.x + threadIdx.x;
  if (idx >= total) return;
  int c = idx % FFN_;
  int row = idx / FFN_;
  float g = up2[(size_t)row * (2 * FFN_) + c];
  float u = up2[(size_t)row * (2 * FFN_) + FFN_ + c];
  float ge = 0.5f * g * (1.f + erff(g * 0.70710678118654752f));
  act[(size_t)row * FFN_ + c] = ge * u;
}

// ---------------- classifier head on last token --------------------------------
__global__ void __launch_bounds__(256)
final_k(const float* __restrict__ xn, const float* __restrict__ fcw,
        const float* __restrict__ fcb, float* __restrict__ out) {
  int wid = blockIdx.x * (blockDim.x >> 5) + (threadIdx.x >> 5);
  if (wid >= BB) return;
  int lane = threadIdx.x & 31;
  const float* xr = xn + ((size_t)wid * SS + (SS - 1)) * EE;
  float s = 0.f;
#pragma unroll
  for (int i = 0; i < 4; ++i) s += xr[lane + 32 * i] * fcw[lane + 32 * i];
#pragma unroll
  for (int o = 16; o >= 1; o >>= 1) s += __shfl_xor(s, o);
  if (lane == 0) out[wid] = s + fcb[0];
}

// ============================ host orchestration ================================
static inline int waveBlocks(long long waves, int wavesPerBlock) {
  return (int)((waves + wavesPerBlock - 1) / wavesPerBlock);
}

extern "C" void kernel_launch(void* const* d_in, const int* in_sizes, int n_in,
                              void* d_out, int out_size, void* d_ws, size_t ws_size,
                              hipStream_t stream) {
  const float* x      = (const float*)d_in[0];
  const float* w_in   = (const float*)d_in[1];
  const float* b_in   = (const float*)d_in[2];
  const float* ln0_w  = (const float*)d_in[3];
  const float* m_up_w = (const float*)d_in[4];
  const float* m_up_b = (const float*)d_in[5];
  const float* m_conv_w = (const float*)d_in[6];
  const float* m_conv_b = (const float*)d_in[7];
  const float* m_q_w  = (const float*)d_in[8];
  const float* m_q_b  = (const float*)d_in[9];
  const float* m_k_w  = (const float*)d_in[10];
  const float* m_k_b  = (const float*)d_in[11];
  const float* m_v_w  = (const float*)d_in[12];
  const float* m_v_b  = (const float*)d_in[13];
  const float* m_ig_w = (const float*)d_in[14];
  const float* m_ig_b = (const float*)d_in[15];
  const float* m_fg_w = (const float*)d_in[16];
  const float* m_fg_b = (const float*)d_in[17];
  const float* m_norm_w = (const float*)d_in[18];
  const float* m_skip = (const float*)d_in[19];
  const float* m_down_w = (const float*)d_in[20];
  const float* m_down_b = (const float*)d_in[21];
  const float* ln1_w  = (const float*)d_in[22];
  const float* s_conv_w = (const float*)d_in[23];
  const float* s_conv_b = (const float*)d_in[24];
  const float* s_gate_w = (const float*)d_in[25];
  const float* s_r    = (const float*)d_in[26];
  const float* s_b    = (const float*)d_in[27];
  const float* s_norm_w = (const float*)d_in[28];
  const float* ln_ff_w = (const float*)d_in[29];
  const float* ff_up_w = (const float*)d_in[30];
  const float* ff_up_b = (const float*)d_in[31];
  const float* ff_down_w = (const float*)d_in[32];
  const float* ff_down_b = (const float*)d_in[33];
  const float* post_w = (const float*)d_in[34];
  const float* fc_w   = (const float*)d_in[35];
  const float* fc_b   = (const float*)d_in[36];
  float* out = (float*)d_out;

  float* ws = (float*)d_ws;
  const size_t M = MM;
  size_t oH  = 0;
  size_t oXN = oH  + M * EE;
  size_t oUP = oXN + M * EE;
  size_t oXC = oUP + M * 2 * DD;
  size_t oQ  = oXC + M * DD;
  size_t oK  = oQ  + M * DD;
  size_t oV  = oK  + M * DD;
  size_t oHC = oV  + M * DD;
  size_t oIG = oHC + M * DD;
  size_t oFG = oIG + (size_t)BH * SS;
  size_t oLF = oFG + (size_t)BH * SS;
  size_t oPM = oLF + (size_t)BH * SS;
  size_t oEM = oPM + (size_t)BH * SS;
  float* bH  = ws + oH;
  float* bXN = ws + oXN;
  float* bUP = ws + oUP;
  float* bXC = ws + oXC;
  float* bQ  = ws + oQ;
  float* bK  = ws + oK;
  float* bV  = ws + oV;
  float* bHC = ws + oHC;
  float* bIG = ws + oIG;
  float* bFG = ws + oFG;
  float* bLF = ws + oLF;
  float* bPM = ws + oPM;
  float* bEM = ws + oEM;
  // reuse (sequenced on the single stream):
  float* bHS  = bV;            // hs after mLSTM (v dead)
  float* bXCS = bK;            // sLSTM conv out (k dead)
  float* bG0  = bUP;           // 4 gates fit exactly in up (z consumed)
  float* bG1  = bUP + M * EE;
  float* bG2  = bUP + 2 * M * EE;
  float* bG3  = bUP + 3 * M * EE;
  float* bYO  = bXC;           // sLSTM outputs (xc dead)
  float* bUP2 = bUP;           // FFN up (gates dead)
  float* bACT = bQ;            // FFN gated act (q dead)

  // 1) h = x @ w_in + b_in
  gemm_wmma_f32<<<waveBlocks((long long)(M / 16) * (EE / 32), 4), 128, 0, stream>>>(
      x, FIN, w_in, EE, b_in, bH, EE, (int)M, EE, FIN, 0);
  // 2) xn = LN(h, ln0)
  layernorm128<<<waveBlocks(M, 8), 256, 0, stream>>>(bH, ln0_w, bXN, (int)M);
  // 3) up = xn @ m_up_w + b
  gemm_wmma_f32<<<waveBlocks((long long)(M / 16) * (2 * DD / 32), 4), 128, 0, stream>>>(
      bXN, EE, m_up_w, 2 * DD, m_up_b, bUP, 2 * DD, (int)M, 2 * DD, EE, 0);
  // 4) xc = silu(causal_conv(x_m))
  {
    int tot = (int)(M * DD);
    convsilu_k<<<(tot + 255) / 256, 256, 0, stream>>>(bUP, 2 * DD, m_conv_w, m_conv_b,
                                                      bXC, DD, tot);
  }
  // 5) q,k from xc; v from x_m
  {
    int tot = (int)(M * DD);
    headwise4_k<<<(tot + 255) / 256, 256, 0, stream>>>(bXC, DD, m_q_w, m_q_b, bQ, tot);
    headwise4_k<<<(tot + 255) / 256, 256, 0, stream>>>(bXC, DD, m_k_w, m_k_b, bK, tot);
    headwise4_k<<<(tot + 255) / 256, 256, 0, stream>>>(bUP, 2 * DD, m_v_w, m_v_b, bV, tot);
  }
  // 6) ig/fg projections (per (b,h,s))
  igfg_k<<<waveBlocks(M, 8), 256, 0, stream>>>(bQ, bK, bV, m_ig_w, m_ig_b, m_fg_w,
                                               m_fg_b, bIG, bFG);
  // 7) per-(b,h) decay scan
  mlstm_scan_k<<<(BH + 255) / 256, 256, 0, stream>>>(bIG, bFG, bLF, bPM, bEM);
  // 8) mLSTM attention (WMMA flash-style), dyn LDS = 2*160*68 + 4*160 + 4*272 floats
  {
    size_t shb = (size_t)(2 * 160 * 68 + 4 * 160 + 4 * 272) * sizeof(float);
    mlstm_attn_k<<<BH, 128, shb, stream>>>(bQ, bK, bV, bLF, bIG, bPM, bEM, bHC);
  }
  // 9) hs = (mhln(hc) + skip*xc) * silu(z)
  fuse_hs_k<<<waveBlocks((long long)M * NHM, 8), 256, 0, stream>>>(bHC, bXC, bUP,
                                                                   m_norm_w, m_skip, bHS);
  // 10) h += hs @ m_down_w + b
  gemm_wmma_f32<<<waveBlocks((long long)(M / 16) * (EE / 32), 4), 128, 0, stream>>>(
      bHS, DD, m_down_w, EE, m_down_b, bH, EE, (int)M, EE, DD, 1);
  // 11) xn = LN(h, ln1)
  layernorm128<<<waveBlocks(M, 8), 256, 0, stream>>>(bH, ln1_w, bXN, (int)M);
  // 12) xcs = silu(causal_conv(xn))
  {
    int tot = (int)(M * EE);
    convsilu_k<<<(tot + 255) / 256, 256, 0, stream>>>(bXN, EE, s_conv_w, s_conv_b,
                                                      bXCS, EE, tot);
  }
  // 13) sLSTM gate projections (i,f from xcs; z,o from xn)
  {
    int tot = (int)(M * EE);
    const int gsz = 4 * 32 * 32;
    headwise32_k<<<(tot + 255) / 256, 256, 0, stream>>>(bXCS, s_gate_w + 0 * gsz, bG0, tot);
    headwise32_k<<<(tot + 255) / 256, 256, 0, stream>>>(bXCS, s_gate_w + 1 * gsz, bG1, tot);
    headwise32_k<<<(tot + 255) / 256, 256, 0, stream>>>(bXN,  s_gate_w + 2 * gsz, bG2, tot);
    headwise32_k<<<(tot + 255) / 256, 256, 0, stream>>>(bXN,  s_gate_w + 3 * gsz, bG3, tot);
  }
  // 14) sequential sLSTM scan (2048 waves)
  slstm_scan_k<<<waveBlocks(BB * 4, 8), 256, 0, stream>>>(bG0, bG1, bG2, bG3, s_r, s_b, bYO);
  // 15) h += mhln(ys)
  fuse_ys_k<<<waveBlocks((long long)M * 4, 8), 256, 0, stream>>>(bYO, s_norm_w, bH);
  // 16) xn = LN(h, ln_ff)
  layernorm128<<<waveBlocks(M, 8), 256, 0, stream>>>(bH, ln_ff_w, bXN, (int)M);
  // 17) up2 = xn @ ff_up_w + b
  gemm_wmma_f32<<<waveBlocks((long long)(M / 16) * (2 * FFN_ / 32), 4), 128, 0, stream>>>(
      bXN, EE, ff_up_w, 2 * FFN_, ff_up_b, bUP2, 2 * FFN_, (int)M, 2 * FFN_, EE, 0);
  // 18) act = gelu(gate) * up
  {
    int tot = (int)(M * FFN_);
    gelumul_k<<<(tot + 255) / 256, 256, 0, stream>>>(bUP2, bACT, tot);
  }
  // 19) h += act @ ff_down_w + b
  gemm_wmma_f32<<<waveBlocks((long long)(M / 16) * (EE / 32), 4), 128, 0, stream>>>(
      bACT, FFN_, ff_down_w, EE, ff_down_b, bH, EE, (int)M, EE, FFN_, 1);
  // 20) xn = LN(h, post_w)
  layernorm128<<<waveBlocks(M, 8), 256, 0, stream>>>(bH, post_w, bXN, (int)M);
  // 21) out[b] = xn[b, S-1, :] @ fc_w + fc_b
  final_k<<<waveBlocks(BB, 8), 256, 0, stream>>>(bXN, fc_w, fc_b, out);
}